// GATGNN_68229850464793
// MI455X (gfx1250) — compile-verified
//
#include <hip/hip_runtime.h>
#include <math.h>

#define NN 50000
#define EDG 800000
#define DD 128
#define GG 64
#define EE (EDG + NN)        // edges + self loops
#define NEG_SLOPE 0.2f

typedef float v2f __attribute__((ext_vector_type(2)));
typedef float v8f __attribute__((ext_vector_type(8)));

// ---------------- order-preserving float<->int encoding for atomicMax -----
__device__ __forceinline__ int enc_f(float f) {
    int i = __float_as_int(f);
    return i >= 0 ? i : (i ^ 0x7FFFFFFF);
}
__device__ __forceinline__ float dec_f(int k) {
    return __int_as_float(k >= 0 ? k : (k ^ 0x7FFFFFFF));
}
#define ENC_NEG_INF 0x807FFFFF   // enc_f(-inf)

// ---------------- WMMA fp32 GEMM: C[N,128] = A[N,128] @ W[128,128] (+bias)
// grid.x = N/16 row-tiles, 256 threads = 8 waves, wave w owns cols [16w,16w+16)
__global__ __launch_bounds__(256) void k_gemm_wmma(
    const float* __restrict__ A, const float* __restrict__ W,
    const float* __restrict__ bias, float* __restrict__ C, int hasBias)
{
    __shared__ float sW[DD * DD];                 // 64 KB of the 320 KB WGP LDS
    for (int i = threadIdx.x; i < DD * DD; i += 256) sW[i] = W[i];
    __syncthreads();

    const int lane = threadIdx.x & 31;
    const int wave = threadIdx.x >> 5;
    const int r0   = blockIdx.x * 16;
    const int c0   = wave * 16;
    const int m    = lane & 15;                   // row within A frag / col within B,C frag
    const int kh   = (lane >> 4) << 1;            // lanes 16-31 hold K+2,K+3

    const float* Arow = A + (size_t)(r0 + m) * DD;
    v8f acc = {};
    for (int kb = 0; kb < DD; kb += 4) {
        v2f a, b;
        a.x = Arow[kb + kh];
        a.y = Arow[kb + kh + 1];
        b.x = sW[(kb + kh) * DD + c0 + m];
        b.y = sW[(kb + kh + 1) * DD + c0 + m];
        acc = __builtin_amdgcn_wmma_f32_16x16x4_f32(
            /*neg_a=*/false, a, /*neg_b=*/false, b,
            /*c_mod=*/(short)0, acc, /*reuse_a=*/false, /*reuse_b=*/false);
    }
    const float bv    = hasBias ? bias[c0 + m] : 0.0f;
    const int   rbase = r0 + ((lane >> 4) << 3); // upper half-lanes hold rows +8
    for (int v = 0; v < 8; ++v)
        C[(size_t)(rbase + v) * DD + c0 + m] = acc[v] + bv;
}

// ---------------- per-row attention logits: wave per row --------------------
__global__ __launch_bounds__(256) void k_alpha(
    const float* __restrict__ h, const float* __restrict__ asrc,
    const float* __restrict__ adst, float* __restrict__ as_, float* __restrict__ ad_)
{
    const int wid  = blockIdx.x * 8 + (threadIdx.x >> 5);
    const int lane = threadIdx.x & 31;
    if (wid >= NN) return;
    float4 hv = ((const float4*)(h + (size_t)wid * DD))[lane];
    float4 sv = ((const float4*)asrc)[lane];
    float4 dv = ((const float4*)adst)[lane];
    float s = hv.x * sv.x + hv.y * sv.y + hv.z * sv.z + hv.w * sv.w;
    float d = hv.x * dv.x + hv.y * dv.y + hv.z * dv.z + hv.w * dv.w;
    for (int off = 16; off > 0; off >>= 1) {
        s += __shfl_xor(s, off, 32);
        d += __shfl_xor(d, off, 32);
    }
    if (lane == 0) { as_[wid] = s; ad_[wid] = d; }
}

// ---------------- reset segment-max / denom / aggregate buffers -------------
__global__ __launch_bounds__(256) void k_init(
    int* __restrict__ mEnc, float* __restrict__ denom, float* __restrict__ agg)
{
    const long i = (long)blockIdx.x * 256 + threadIdx.x;
    if (i < NN) { mEnc[i] = ENC_NEG_INF; denom[i] = 0.0f; }
    if (i < (long)NN * DD) agg[i] = 0.0f;
}

__device__ __forceinline__ void edge_sd(int e, const int* src, const int* dst,
                                        int& s, int& d) {
    if (e < EDG) { s = src[e]; d = dst[e]; }
    else         { s = e - EDG; d = s; }          // self loop
}

__device__ __forceinline__ float edge_logit(int s, int d,
    const float* __restrict__ as_, const float* __restrict__ ad_) {
    float v = as_[s] + ad_[d];
    return v > 0.0f ? v : NEG_SLOPE * v;          // leaky relu
}

// ---------------- segment max over destination ------------------------------
__global__ __launch_bounds__(256) void k_edge_max(
    const int* __restrict__ src, const int* __restrict__ dst,
    const float* __restrict__ as_, const float* __restrict__ ad_,
    int* __restrict__ mEnc)
{
    const int e = blockIdx.x * 256 + threadIdx.x;
    if (e >= EE) return;
    int s, d; edge_sd(e, src, dst, s, d);
    atomicMax(mEnc + d, enc_f(edge_logit(s, d, as_, ad_)));
}

// ---------------- exp(e - max) and segment sum ------------------------------
__global__ __launch_bounds__(256) void k_edge_expsum(
    const int* __restrict__ src, const int* __restrict__ dst,
    const float* __restrict__ as_, const float* __restrict__ ad_,
    const int* __restrict__ mEnc, float* __restrict__ denom,
    float* __restrict__ wbuf)
{
    const int e = blockIdx.x * 256 + threadIdx.x;
    if (e >= EE) return;
    int s, d; edge_sd(e, src, dst, s, d);
    float v = edge_logit(s, d, as_, ad_);
    float w = __expf(v - dec_f(mEnc[d]));
    wbuf[e] = w;
    unsafeAtomicAdd(denom + d, w);
}

// ---------------- weighted scatter: agg[dst] += alpha * h[src], wave/edge ---
__global__ __launch_bounds__(256) void k_edge_scatter(
    const int* __restrict__ src, const int* __restrict__ dst,
    const float* __restrict__ wbuf, const float* __restrict__ denom,
    const float* __restrict__ h, float* __restrict__ agg)
{
    const int wid  = blockIdx.x * 8 + (threadIdx.x >> 5);
    const int lane = threadIdx.x & 31;
    if (wid >= EE) return;
    int s, d; edge_sd(wid, src, dst, s, d);
    const float alpha = wbuf[wid] / denom[d];
    float4 hv = ((const float4*)(h + (size_t)s * DD))[lane];
    float* out = agg + (size_t)d * DD + lane * 4;
    unsafeAtomicAdd(out + 0, alpha * hv.x);
    unsafeAtomicAdd(out + 1, alpha * hv.y);
    unsafeAtomicAdd(out + 2, alpha * hv.z);
    unsafeAtomicAdd(out + 3, alpha * hv.w);
}

// ---------------- out = relu(agg + bias) ------------------------------------
__global__ __launch_bounds__(256) void k_bias_relu(
    const float* __restrict__ agg, const float* __restrict__ bias,
    float* __restrict__ out)
{
    const long i = (long)blockIdx.x * 256 + threadIdx.x;
    if (i >= (long)NN * DD) return;
    float c = agg[i] + bias[i & (DD - 1)];
    out[i] = c > 0.0f ? c : 0.0f;
}

// ---------------- mean pool: zero + wave-per-node accumulate ----------------
__global__ __launch_bounds__(256) void k_pool_zero(float* __restrict__ pooled,
                                                   float* __restrict__ cnt)
{
    const int i = blockIdx.x * 256 + threadIdx.x;
    if (i < GG * DD) pooled[i] = 0.0f;
    if (i < GG) cnt[i] = 0.0f;
}

__global__ __launch_bounds__(256) void k_pool(
    const float* __restrict__ h, const int* __restrict__ batch,
    float* __restrict__ pooled, float* __restrict__ cnt)
{
    const int wid  = blockIdx.x * 8 + (threadIdx.x >> 5);
    const int lane = threadIdx.x & 31;
    if (wid >= NN) return;
    const int g = batch[wid];
    float4 hv = ((const float4*)(h + (size_t)wid * DD))[lane];
    float* p = pooled + (size_t)g * DD + lane * 4;
    unsafeAtomicAdd(p + 0, hv.x);
    unsafeAtomicAdd(p + 1, hv.y);
    unsafeAtomicAdd(p + 2, hv.z);
    unsafeAtomicAdd(p + 3, hv.w);
    if (lane == 0) unsafeAtomicAdd(cnt + g, 1.0f);
}

// ---------------- final GEMV: out[g] = (pooled[g]/cnt[g]) . out_w + out_b ---
__global__ __launch_bounds__(256) void k_final(
    const float* __restrict__ pooled, const float* __restrict__ cnt,
    const float* __restrict__ out_w, const float* __restrict__ out_b,
    float* __restrict__ out)
{
    const int wid  = blockIdx.x * 8 + (threadIdx.x >> 5);
    const int lane = threadIdx.x & 31;
    if (wid >= GG) return;
    float4 pv = ((const float4*)(pooled + (size_t)wid * DD))[lane];
    float4 wv = ((const float4*)out_w)[lane];
    float s = pv.x * wv.x + pv.y * wv.y + pv.z * wv.z + pv.w * wv.w;
    for (int off = 16; off > 0; off >>= 1) s += __shfl_xor(s, off, 32);
    if (lane == 0) out[wid] = s / cnt[wid] + out_b[0];
}

// ===========================================================================
extern "C" void kernel_launch(void* const* d_in, const int* in_sizes, int n_in,
                              void* d_out, int out_size, void* d_ws, size_t ws_size,
                              hipStream_t stream)
{
    const float* x      = (const float*)d_in[0];
    const int*   ei     = (const int*)  d_in[1];
    const int*   batch  = (const int*)  d_in[2];
    const float* lin_w  = (const float*)d_in[3];
    const float* lin_b  = (const float*)d_in[4];
    const float* w1     = (const float*)d_in[5];
    const float* asrc1  = (const float*)d_in[6];
    const float* adst1  = (const float*)d_in[7];
    const float* b1     = (const float*)d_in[8];
    const float* w2     = (const float*)d_in[9];
    const float* asrc2  = (const float*)d_in[10];
    const float* adst2  = (const float*)d_in[11];
    const float* b2     = (const float*)d_in[12];
    const float* out_w  = (const float*)d_in[13];
    const float* out_b  = (const float*)d_in[14];
    float*       out    = (float*)d_out;

    const int* src = ei;           // edge_index[0]
    const int* dst = ei + EDG;     // edge_index[1]

    // ---- workspace carve-up (floats) ----
    float* hA     = (float*)d_ws;          // [N,D]  layer input
    float* hB     = hA + (size_t)NN * DD;  // [N,D]  projected h
    float* hC     = hB + (size_t)NN * DD;  // [N,D]  aggregate
    float* as_    = hC + (size_t)NN * DD;  // [N]
    float* ad_    = as_ + NN;              // [N]
    float* denom  = ad_ + NN;              // [N]
    int*   mEnc   = (int*)(denom + NN);    // [N]
    float* wbuf   = (float*)(mEnc + NN);   // [E+N]
    float* pooled = wbuf + EE;             // [G,D]
    float* cnt    = pooled + GG * DD;      // [G]

    const dim3 blk(256);
    const dim3 gGemm(NN / 16);                       // 3125 row tiles
    const dim3 gRowWave((NN + 7) / 8);               // wave per row
    const dim3 gEdge((EE + 255) / 256);              // thread per edge
    const dim3 gEdgeWave((EE + 7) / 8);              // wave per edge
    const dim3 gND(((long)NN * DD + 255) / 256);     // thread per element

    // h = x @ lin_in_w + lin_in_b
    k_gemm_wmma<<<gGemm, blk, 0, stream>>>(x, lin_w, lin_b, hA, 1);

    // ---------- GAT layer 1 ----------
    k_gemm_wmma<<<gGemm, blk, 0, stream>>>(hA, w1, nullptr, hB, 0);
    k_alpha<<<gRowWave, blk, 0, stream>>>(hB, asrc1, adst1, as_, ad_);
    k_init<<<gND, blk, 0, stream>>>(mEnc, denom, hC);
    k_edge_max<<<gEdge, blk, 0, stream>>>(src, dst, as_, ad_, mEnc);
    k_edge_expsum<<<gEdge, blk, 0, stream>>>(src, dst, as_, ad_, mEnc, denom, wbuf);
    k_edge_scatter<<<gEdgeWave, blk, 0, stream>>>(src, dst, wbuf, denom, hB, hC);
    k_bias_relu<<<gND, blk, 0, stream>>>(hC, b1, hA);

    // ---------- GAT layer 2 ----------
    k_gemm_wmma<<<gGemm, blk, 0, stream>>>(hA, w2, nullptr, hB, 0);
    k_alpha<<<gRowWave, blk, 0, stream>>>(hB, asrc2, adst2, as_, ad_);
    k_init<<<gND, blk, 0, stream>>>(mEnc, denom, hC);
    k_edge_max<<<gEdge, blk, 0, stream>>>(src, dst, as_, ad_, mEnc);
    k_edge_expsum<<<gEdge, blk, 0, stream>>>(src, dst, as_, ad_, mEnc, denom, wbuf);
    k_edge_scatter<<<gEdgeWave, blk, 0, stream>>>(src, dst, wbuf, denom, hB, hC);
    k_bias_relu<<<gND, blk, 0, stream>>>(hC, b2, hA);

    // ---------- mean pool + output head ----------
    k_pool_zero<<<dim3((GG * DD + 255) / 256), blk, 0, stream>>>(pooled, cnt);
    k_pool<<<gRowWave, blk, 0, stream>>>(hA, batch, pooled, cnt);
    k_final<<<dim3((GG + 7) / 8), blk, 0, stream>>>(pooled, cnt, out_w, out_b, out);
}